// LiftSplatShoot_33655363732173
// MI455X (gfx1250) — compile-verified
//
#include <hip/hip_runtime.h>
#include <hip/hip_bf16.h>
#include <stdint.h>

// ---- problem constants (match reference) ----
#define NB   8
#define ND   41
#define NH   32
#define NW   88
#define NC   64
#define NXX  200   // x grid
#define NZZ  200   // z grid
#define PPB  128   // points per block (128 divides 115456 exactly -> no batch straddle)
#define PT_PER_BATCH (ND * NH * NW)      // 115456
#define NPOINTS      (NB * PT_PER_BATCH) // 923648
#define OUT_ELEMS    (NB * NC * NZZ * NXX) // 20,480,000 floats (same count as pooled [320000,64])

// ---------------- CDNA5 async global->LDS DMA (inline asm, per CDNA5 ISA) ----
// LDS address is the LDS base plus the VDST operand; vaddr = 64-bit global address (GV mode).
__device__ __forceinline__ void async_ld16(uint32_t lds_byte_off, const float* gaddr) {
    asm volatile("global_load_async_to_lds_b128 %0, %1, off"
                 :: "v"(lds_byte_off), "v"(gaddr)
                 : "memory");
}
__device__ __forceinline__ void wait_async0() {
    asm volatile("s_wait_asynccnt 0x0" ::: "memory");
}

__device__ __forceinline__ void atomic_add_f32(float* p, float v) {
    __hip_atomic_fetch_add(p, v, __ATOMIC_RELAXED, __HIP_MEMORY_SCOPE_AGENT);
}

// ---------------- lowering probe: FIRST function in the module so the disasm snippet
// shows whether __hip_atomic_fetch_add becomes global_atomic_add_f32 or a CAS loop.
// (Never launched; kernels are emitted regardless of host references.)
__global__ void aaa_atomic_probe(float* p, float v) {
    atomic_add_f32(p + threadIdx.x, v);
}

// ---------------- small helpers ----------------
__device__ __forceinline__ void inv3x3(const float* __restrict__ m, float* o) {
    float a = m[0], b = m[1], c = m[2];
    float d = m[3], e = m[4], f = m[5];
    float g = m[6], h = m[7], i = m[8];
    float A  =  (e * i - f * h);
    float Bc = -(d * i - f * g);
    float Cc =  (d * h - e * g);
    float det = a * A + b * Bc + c * Cc;
    float r = 1.0f / det;
    o[0] = A * r;              o[1] = (c * h - b * i) * r; o[2] = (b * f - c * e) * r;
    o[3] = Bc * r;             o[4] = (a * i - c * g) * r; o[5] = (c * d - a * f) * r;
    o[6] = Cc * r;             o[7] = (b * g - a * h) * r; o[8] = (a * e - b * d) * r;
}

// ---------------- main splat kernel ----------------
// DIRECT=true : accumulate straight into out[b, c, z, x]   (stride 40000 between channels)
// DIRECT=false: accumulate into pooled[(x*200+z)*8+b, c]   (coalesced, transposed later)
template <bool DIRECT>
__global__ __launch_bounds__(PPB) void splat_kernel(
        const float* __restrict__ x,
        const float* __restrict__ intrins,
        const float* __restrict__ post_rots,
        const float* __restrict__ post_trans,
        float* __restrict__ dst)
{
    __shared__ __align__(16) float feat[PPB * NC]; // 32 KB feature tile
    __shared__ int sCX[PPB];
    __shared__ int sCZ[PPB];

    const int t    = threadIdx.x;
    const int pid0 = blockIdx.x * PPB;            // first point of this block
    const int b    = pid0 / PT_PER_BATCH;         // block-uniform batch index

    // ---- kick off async DMA of this block's 128x64 f32 feature tile into LDS ----
    const float*  gsrc    = x + (size_t)pid0 * NC;
    const uint32_t ldsOff = (uint32_t)(uintptr_t)(&feat[0]); // low 32 bits == LDS offset (ISA 10.2)
    #pragma unroll
    for (int it = 0; it < (PPB * NC) / (4 * PPB); ++it) {    // 16 x b128 per thread
        int f4 = it * PPB + t;                               // float4 index within tile
        async_ld16(ldsOff + (uint32_t)f4 * 16u, gsrc + (size_t)f4 * 4);
    }

    // ---- geometry for this thread's point (overlaps the DMA) ----
    const int pid = pid0 + t;
    const int w   = pid % NW;
    const int r1  = pid / NW;
    const int h   = r1 % NH;
    const int r2  = r1 / NH;
    const int d   = r2 % ND;

    float Ri[9], Ki[9];
    inv3x3(post_rots + b * 9, Ri);
    inv3x3(intrins   + b * 9, Ki);
    const float tx = post_trans[b * 3 + 0];
    const float ty = post_trans[b * 3 + 1];
    const float tz = post_trans[b * 3 + 2];

    // frustum (u, v, depth): linspace(0,1407,88), linspace(0,511,32), 4..44
    const float u   = (float)w * (1407.0f / 87.0f);
    const float v   = (float)h * (511.0f / 31.0f);
    const float dep = 4.0f + (float)d;

    float px = u - tx, py = v - ty, pz = dep - tz;
    float qx = Ri[0] * px + Ri[1] * py + Ri[2] * pz;
    float qy = Ri[3] * px + Ri[4] * py + Ri[5] * pz;
    float qz = Ri[6] * px + Ri[7] * py + Ri[8] * pz;
    float rx = qx * qz, ry = qy * qz, rz = qz;
    float gx = Ki[0] * rx + Ki[1] * ry + Ki[2] * rz;
    float gy = Ki[3] * rx + Ki[4] * ry + Ki[5] * rz;
    float gz = Ki[6] * rx + Ki[7] * ry + Ki[8] * rz;

    // coords = trunc((geom - (BX - DX/2)) / DX); trunc-toward-zero matches astype(int32)
    int cx = (int)((gx + 50.0f) / 0.5f);
    int cy = (int)((gy + 10.0f) / 20.0f);
    int cz = (int)((gz + 50.0f) / 0.5f);
    bool kept = (cx >= 0) & (cx < NXX) & (cy == 0) & (cz >= 0) & (cz < NZZ);
    sCX[t] = kept ? cx : -1;
    sCZ[t] = cz;

    wait_async0();
    __syncthreads();

    // ---- warp-per-point scatter: lane handles channels (lane, lane+32) ----
    const int lane = t & 31;
    const int wrp  = t >> 5;
    #pragma unroll 1
    for (int i = 0; i < 32; ++i) {
        int p  = (wrp << 5) + i;
        int vx = sCX[p];
        if (vx < 0) continue;
        int vz = sCZ[p];
        float f0 = feat[p * NC + lane];        // bank-conflict-free (bank == lane)
        float f1 = feat[p * NC + lane + 32];
        if (DIRECT) {
            size_t base = (size_t)b * (NC * NZZ * NXX) + (size_t)vz * NXX + (size_t)vx;
            atomic_add_f32(&dst[base + (size_t)lane * (NZZ * NXX)], f0);
            atomic_add_f32(&dst[base + (size_t)(lane + 32) * (NZZ * NXX)], f1);
        } else {
            size_t base = ((size_t)(vx * NZZ + vz) * NB + (size_t)b) * NC;
            atomic_add_f32(&dst[base + lane], f0);       // coalesced 128B atomic burst
            atomic_add_f32(&dst[base + lane + 32], f1);
        }
    }
}

// ---------------- bulk kernels ----------------
__global__ __launch_bounds__(256) void zero_kernel(float4* __restrict__ p, int n4) {
    int i = blockIdx.x * 256 + threadIdx.x;
    if (i < n4) {
        float4 z; z.x = 0.f; z.y = 0.f; z.z = 0.f; z.w = 0.f;
        p[i] = z;
    }
}

// pooled[(x*200+z)*8+b, c]  ->  out[((b*64+c)*200+z)*200+x]
__global__ __launch_bounds__(256) void transpose_kernel(const float* __restrict__ pooled,
                                                        float* __restrict__ out)
{
    int o = blockIdx.x * 256 + threadIdx.x;
    if (o >= OUT_ELEMS) return;
    int xx = o % NXX;
    int t1 = o / NXX;
    int z  = t1 % NZZ;
    int t2 = t1 / NZZ;
    int c  = t2 % NC;
    int b  = t2 / NC;
    out[o] = pooled[(((size_t)xx * NZZ + z) * NB + b) * NC + c];
}

// ---------------- launcher ----------------
extern "C" void kernel_launch(void* const* d_in, const int* in_sizes, int n_in,
                              void* d_out, int out_size, void* d_ws, size_t ws_size,
                              hipStream_t stream) {
    (void)in_sizes; (void)n_in; (void)out_size;
    const float* x          = (const float*)d_in[0];
    const float* intrins    = (const float*)d_in[1];
    const float* post_rots  = (const float*)d_in[2];
    const float* post_trans = (const float*)d_in[3];
    float* out = (float*)d_out;

    const int n4        = OUT_ELEMS / 4;                // 5,120,000
    const int zeroGrid  = n4 / 256;                     // 20,000 (exact)
    const int splatGrid = NPOINTS / PPB;                // 7,216 (exact)

    const bool pooledPath = ws_size >= (size_t)OUT_ELEMS * sizeof(float);
    if (pooledPath) {
        float* pooled = (float*)d_ws;
        zero_kernel<<<zeroGrid, 256, 0, stream>>>((float4*)pooled, n4);
        splat_kernel<false><<<splatGrid, PPB, 0, stream>>>(x, intrins, post_rots, post_trans, pooled);
        transpose_kernel<<<OUT_ELEMS / 256, 256, 0, stream>>>(pooled, out);
    } else {
        zero_kernel<<<zeroGrid, 256, 0, stream>>>((float4*)out, n4);
        splat_kernel<true><<<splatGrid, PPB, 0, stream>>>(x, intrins, post_rots, post_trans, out);
    }
}